// GATLayer_58007828300548
// MI455X (gfx1250) — compile-verified
//
#include <hip/hip_runtime.h>
#include <cstddef>

#define NTOT   65536      // B*N nodes
#define IN_F   128
#define OUT_F  128
#define EDGES  1048576
#define NEG_SLOPE 0.01f

typedef __attribute__((ext_vector_type(2))) float v2f;
typedef __attribute__((ext_vector_type(8))) float v8f;

// ---- monotonic float<->uint encoding for atomic max over signed floats ----
__device__ __forceinline__ unsigned enc_f32(float f) {
    unsigned b = __float_as_uint(f);
    return (b & 0x80000000u) ? ~b : (b | 0x80000000u);
}
__device__ __forceinline__ float dec_f32(unsigned k) {
    return (k & 0x80000000u) ? __uint_as_float(k ^ 0x80000000u)
                             : __uint_as_float(~k);
}

// ---------------------------------------------------------------------------
// Kernel 0: zero a u32 range (used for out, emax, denom)
// ---------------------------------------------------------------------------
__global__ void zero_u32(unsigned* __restrict__ p, size_t n) {
    size_t i = (size_t)blockIdx.x * blockDim.x + threadIdx.x;
    size_t stride = (size_t)gridDim.x * blockDim.x;
    for (; i < n; i += stride) p[i] = 0u;
}

// ---------------------------------------------------------------------------
// Kernel 1: z = h @ fc_w^T via V_WMMA_F32_16X16X4_F32, fused attention scores
// One wave computes a 16(M) x 128(N) tile of z; fc_w staged in LDS (64 KB).
// A-frag (16x4 f32): lane L holds h[row0+L%16][kb..kb+1], kb = 4*s + 2*(L/16)
// B-frag (4x16 f32): lane L holds fc_w[16t+L%16][kb..kb+1]  (B[k][n]=fc_w[n][k])
// C/D (16x16 f32):   VGPR r, lane L -> row r+8*(L/16), col L%16 (per n-tile)
// ---------------------------------------------------------------------------
__global__ __launch_bounds__(256) void gat_gemm_wmma(
    const float* __restrict__ h, const float* __restrict__ fc_w,
    const float* __restrict__ attn_w,
    float* __restrict__ z, float* __restrict__ s_src, float* __restrict__ s_dst)
{
    __shared__ float sB[OUT_F * IN_F];   // 64 KB of fc_w, row-major
    for (int i = threadIdx.x; i < (OUT_F * IN_F) / 4; i += blockDim.x)
        ((float4*)sB)[i] = ((const float4*)fc_w)[i];
    __syncthreads();

    const int wave = threadIdx.x >> 5;
    const int lane = threadIdx.x & 31;
    const int half = lane >> 4;          // 0: lanes 0-15, 1: lanes 16-31
    const int lm   = lane & 15;
    const int row0 = (blockIdx.x * 8 + wave) * 16;

    v8f acc[8];
    const v8f vzero = {0.f,0.f,0.f,0.f,0.f,0.f,0.f,0.f};
#pragma unroll
    for (int t = 0; t < 8; ++t) acc[t] = vzero;

    const float* hrow = h + (size_t)(row0 + lm) * IN_F;
#pragma unroll 4
    for (int s = 0; s < 32; ++s) {
        const int kb = 4 * s + 2 * half;
        v2f a = *(const v2f*)(hrow + kb);
#pragma unroll
        for (int t = 0; t < 8; ++t) {
            v2f b = *(const v2f*)(&sB[(16 * t + lm) * IN_F + kb]);
            acc[t] = __builtin_amdgcn_wmma_f32_16x16x4_f32(
                false, a, false, b, (short)0, acc[t], false, false);
        }
    }

    // Fused epilogue: per-node attention dot products
    float ps[8], pd[8];
#pragma unroll
    for (int r = 0; r < 8; ++r) { ps[r] = 0.f; pd[r] = 0.f; }
#pragma unroll
    for (int t = 0; t < 8; ++t) {
        const float wsrc = attn_w[16 * t + lm];
        const float wdst = attn_w[OUT_F + 16 * t + lm];
#pragma unroll
        for (int r = 0; r < 8; ++r) {
            ps[r] += acc[t][r] * wsrc;
            pd[r] += acc[t][r] * wdst;
        }
    }
    // reduce across the 16 lanes of each half (wave32; halves stay disjoint)
#pragma unroll
    for (int m = 1; m <= 8; m <<= 1) {
#pragma unroll
        for (int r = 0; r < 8; ++r) {
            ps[r] += __shfl_xor(ps[r], m, 32);
            pd[r] += __shfl_xor(pd[r], m, 32);
        }
    }

    // store z (coalesced across lm) and the per-row scores
#pragma unroll
    for (int t = 0; t < 8; ++t)
#pragma unroll
        for (int r = 0; r < 8; ++r)
            z[(size_t)(row0 + r + 8 * half) * OUT_F + 16 * t + lm] = acc[t][r];

    if (lm == 0) {
#pragma unroll
        for (int r = 0; r < 8; ++r) {
            const int row = row0 + r + 8 * half;
            s_src[row] = ps[r];
            s_dst[row] = pd[r];
        }
    }
}

// ---------------------------------------------------------------------------
// Kernel 2: per-edge score + leaky_relu, segment max via native u32 atomic max
// ---------------------------------------------------------------------------
__global__ __launch_bounds__(256) void edge_score(
    const int* __restrict__ src, const int* __restrict__ dst,
    const float* __restrict__ s_src, const float* __restrict__ s_dst,
    float* __restrict__ escore, unsigned* __restrict__ emax)
{
    int e = blockIdx.x * blockDim.x + threadIdx.x;
    if (e >= EDGES) return;
    const int d = dst[e];
    float v = s_src[src[e]] + s_dst[d];
    v = (v > 0.f) ? v : NEG_SLOPE * v;
    escore[e] = v;
    atomicMax(&emax[d], enc_f32(v));
}

// ---------------------------------------------------------------------------
// Kernel 3: ex = exp(e - max), segment sum via hw f32 atomic add (L2-resident)
// ---------------------------------------------------------------------------
__global__ __launch_bounds__(256) void edge_exp(
    const int* __restrict__ dst, float* __restrict__ escore,
    const unsigned* __restrict__ emax, float* __restrict__ denom)
{
    int e = blockIdx.x * blockDim.x + threadIdx.x;
    if (e >= EDGES) return;
    const int d = dst[e];
    const float ex = __expf(escore[e] - dec_f32(emax[d]));
    escore[e] = ex;
    unsafeAtomicAdd(&denom[d], ex);
}

// ---------------------------------------------------------------------------
// Kernel 4: out[dst] += alpha * z[src]; one wave per edge, float4 per lane,
// hw f32 atomics (out + z both fit in the 192 MB L2)
// ---------------------------------------------------------------------------
__global__ __launch_bounds__(256) void edge_scatter(
    const int* __restrict__ src, const int* __restrict__ dst,
    const float* __restrict__ z, const float* __restrict__ escore,
    const float* __restrict__ denom, float* __restrict__ out)
{
    const int gid  = blockIdx.x * blockDim.x + threadIdx.x;
    const int e    = gid >> 5;
    const int lane = gid & 31;
    if (e >= EDGES) return;
    const int sN = src[e], dN = dst[e];
    const float alpha = escore[e] / denom[dN];
    const float4 zv = *(const float4*)(z + (size_t)sN * OUT_F + lane * 4);
    float* op = out + (size_t)dN * OUT_F + lane * 4;
    unsafeAtomicAdd(op + 0, alpha * zv.x);
    unsafeAtomicAdd(op + 1, alpha * zv.y);
    unsafeAtomicAdd(op + 2, alpha * zv.z);
    unsafeAtomicAdd(op + 3, alpha * zv.w);
}

// ---------------------------------------------------------------------------
extern "C" void kernel_launch(void* const* d_in, const int* in_sizes, int n_in,
                              void* d_out, int out_size, void* d_ws, size_t ws_size,
                              hipStream_t stream)
{
    const float* h      = (const float*)d_in[0];
    const float* fc_w   = (const float*)d_in[1];
    const float* attn_w = (const float*)d_in[2];
    const int*   src    = (const int*)d_in[3];
    const int*   dst    = (const int*)d_in[4];
    float*       out    = (float*)d_out;

    // workspace carve-up (~38.8 MB)
    float*    z      = (float*)d_ws;
    float*    s_src  = z + (size_t)NTOT * OUT_F;
    float*    s_dst  = s_src + NTOT;
    unsigned* emax   = (unsigned*)(s_dst + NTOT);
    float*    denom  = (float*)emax + NTOT;
    float*    escore = denom + NTOT;

    // zero out, emax (encoded init 0 == below any real score), denom
    zero_u32<<<4096, 256, 0, stream>>>((unsigned*)out, (size_t)NTOT * OUT_F);
    zero_u32<<<512, 256, 0, stream>>>(emax, (size_t)2 * NTOT);

    // GEMM + fused node scores: 512 blocks x 8 waves x 16 rows = 65536 rows
    gat_gemm_wmma<<<512, 256, 0, stream>>>(h, fc_w, attn_w, z, s_src, s_dst);

    edge_score<<<EDGES / 256, 256, 0, stream>>>(src, dst, s_src, s_dst, escore, emax);
    edge_exp<<<EDGES / 256, 256, 0, stream>>>(dst, escore, emax, denom);
    edge_scatter<<<(EDGES * 32) / 256, 256, 0, stream>>>(src, dst, z, escore, denom, out);
}